// BEVDetectionLoss_80101140070751
// MI455X (gfx1250) — compile-verified
//
#include <hip/hip_runtime.h>
#include <stdint.h>

#define BB      4
#define NCELLS  (1 << 20)
#define TOTAL   (BB * NCELLS)      // 4194304
#define NBOX    128
#define NCLS    10
#define BD      7
#define THR     256
#define NBLK    1024
#define TILE    4096               // floats per TDM tile (16 KB)

// ---- workspace layout (bytes) ----
#define OFF_HIST0   0u             // 4096 u32
#define OFF_HIST1   16384u         // 4096 u32
#define OFF_HIST2   32768u         // 256 u32
#define OFF_POSCNT  33792u         // 4 i32
#define OFF_POSCELL 33808u         // 512 i32
#define OFF_BSUMS   35856u         // 12 f32: cls[4], box[4], objpos[4]
#define OFF_SCAL    35904u         // 8 u32: P,k,T0,r0,prefix01,r1,u_star,r
#define OFF_PART    35968u         // NBLK f32 block partials
#define WS_BYTES    (OFF_PART + NBLK * 4u)

#if defined(__AMDGCN__) && __has_builtin(__builtin_amdgcn_tensor_load_to_lds)
#define HAVE_TDM 1
typedef unsigned int u32x4 __attribute__((ext_vector_type(4)));
typedef int          i32x4 __attribute__((ext_vector_type(4)));
typedef int          i32x8 __attribute__((ext_vector_type(8)));
#else
#define HAVE_TDM 0
#endif

__device__ __forceinline__ uint32_t fkey(float x) {
  uint32_t b = __float_as_uint(x);
  return b ^ ((b & 0x80000000u) ? 0xFFFFFFFFu : 0x80000000u);
}
__device__ __forceinline__ float fkey_inv(uint32_t u) {
  uint32_t b = (u & 0x80000000u) ? (u ^ 0x80000000u) : ~u;
  return __uint_as_float(b);
}
__device__ __forceinline__ float softplusf(float z) {
  return (z > 0.0f) ? (z + log1pf(expf(-z))) : log1pf(expf(z));
}

#if HAVE_TDM
// Issue one Tensor-Data-Mover DMA: TILE fp32 elements, global -> LDS.
// Must be executed from a wave-uniform context (we call it from thread 0).
__device__ __forceinline__ void tdm_load_tile(const float* gsrc, uint32_t lds_off) {
  const uint64_t ga = (uint64_t)(uintptr_t)gsrc;
  u32x4 g0;
  g0[0] = 1u;                                                // count=1, user mode
  g0[1] = lds_off;                                           // lds_addr
  g0[2] = (uint32_t)(ga & 0xFFFFFFFFu);                      // global_addr[31:0]
  g0[3] = (uint32_t)((ga >> 32) & 0x01FFFFFFu) | (2u << 30); // addr[56:32] | type=2
  i32x8 g1;
  g1[0] = (int)(2u << 16);                                   // data_size=4B, wg_mask=0
  g1[1] = (int)(((uint32_t)TILE & 0xFFFFu) << 16);           // tensor_dim0 lo16
  g1[2] = (int)((((uint32_t)TILE >> 16) & 0xFFFFu) | (1u << 16)); // dim0 hi | tensor_dim1=1 lo
  g1[3] = (int)(((uint32_t)TILE & 0xFFFFu) << 16);           // tensor_dim1 hi=0 | tile_dim0
  g1[4] = 1;                                                 // tile_dim1=1, tile_dim2=0
  g1[5] = (int)TILE;                                         // tensor_dim0_stride lo32
  g1[6] = 0;
  g1[7] = 0;
  i32x4 z4 = {0, 0, 0, 0};
#if defined(__clang_major__) && (__clang_major__ >= 23)
  i32x8 z8 = {0, 0, 0, 0, 0, 0, 0, 0};
  __builtin_amdgcn_tensor_load_to_lds(g0, g1, z4, z4, z8, 0);
#else
  __builtin_amdgcn_tensor_load_to_lds(g0, g1, z4, z4, 0);
#endif
}
// s_wait_tensorcnt takes an immediate -> must be a literal at the call site.
#if __has_builtin(__builtin_amdgcn_s_wait_tensorcnt)
#define TDM_WAIT(n) __builtin_amdgcn_s_wait_tensorcnt(n)
#else
#define TDM_WAIT(n) asm volatile("s_wait_tensorcnt " #n ::: "memory")
#endif
#endif  // HAVE_TDM

// ---------------- K2: histogram level 0 (top 12 bits of order key) ----------------
// Double-buffered TDM pipeline: issue both tile DMAs first (32 KB in flight
// while the block zeroes its LDS histogram), then use in-order TENSORcnt
// (wait<=1 -> first tile landed) to overlap DMA of tile B with histogramming
// of tile A. 512 blocks x 2 tiles x 4096 floats = 4.19M.
__global__ void k_hist0(const float* __restrict__ obj, uint8_t* __restrict__ ws) {
  __shared__ float    s_data[2][TILE];
  __shared__ uint32_t s_hist[4096];
  const float* gsrc = obj + (size_t)blockIdx.x * (2 * TILE);
#if HAVE_TDM
  if (threadIdx.x == 0) {   // DMA issue first: data moves while we zero s_hist
    tdm_load_tile(gsrc,        (uint32_t)(uintptr_t)(&s_data[0][0]));
    tdm_load_tile(gsrc + TILE, (uint32_t)(uintptr_t)(&s_data[1][0]));
  }
  for (int i = threadIdx.x; i < 4096; i += THR) s_hist[i] = 0u;
  if (threadIdx.x == 0) TDM_WAIT(1);   // in-order: tile 0 in LDS, tile 1 in flight
  asm volatile("" ::: "memory");       // TDM wrote LDS behind the compiler's back
  __syncthreads();
  for (int i = threadIdx.x; i < TILE; i += THR) {
    uint32_t u = fkey(s_data[0][i]);
    atomicAdd(&s_hist[u >> 20], 1u);
  }
  __syncthreads();
  if (threadIdx.x == 0) TDM_WAIT(0);   // tile 1 landed
  asm volatile("" ::: "memory");
  __syncthreads();
  for (int i = threadIdx.x; i < TILE; i += THR) {
    uint32_t u = fkey(s_data[1][i]);
    atomicAdd(&s_hist[u >> 20], 1u);
  }
#else
  for (int i = threadIdx.x; i < 4096; i += THR) s_hist[i] = 0u;
  __syncthreads();
  for (int i = threadIdx.x; i < 2 * TILE; i += THR) {
    uint32_t u = fkey(gsrc[i]);
    atomicAdd(&s_hist[u >> 20], 1u);
  }
#endif
  __syncthreads();
  uint32_t* gh = (uint32_t*)(ws + OFF_HIST0);
  for (int i = threadIdx.x; i < 4096; i += THR) {
    uint32_t c = s_hist[i];
    if (c) atomicAdd(&gh[i], c);
  }
}

// ---------------- K0: zero workspace ----------------
__global__ void k_zero(uint32_t* __restrict__ w, int n) {
  int i = blockIdx.x * blockDim.x + threadIdx.x;
  if (i < n) w[i] = 0u;
}

// ---------------- K1: GT assignment + positive-cell losses ----------------
__global__ void k_assign(const float* __restrict__ cls,
                         const float* __restrict__ obj,
                         const float* __restrict__ boxp,
                         const float* __restrict__ gtb,
                         const int*   __restrict__ gtl,
                         const float* __restrict__ gtm,
                         uint8_t* __restrict__ ws) {
  __shared__ int   s_cell[NBOX];
  __shared__ int   s_pos[NBOX];
  __shared__ float s_c[NBOX], s_b[NBOX], s_o[NBOX];
  const int b = blockIdx.x, t = threadIdx.x;
  const float* box = gtb + ((size_t)b * NBOX + t) * BD;
  const float x = box[0], y = box[1];
  const int   lab = gtl[b * NBOX + t];
  const float msk = gtm[b * NBOX + t];
  bool valid = (msk > 0.5f) && (lab >= 0) &&
               (x >= -51.2f) && (x <= 51.2f) && (y >= -51.2f) && (y <= 51.2f);
  int gx = (int)floorf((x - (-51.2f)) / 0.1f);
  int gy = (int)floorf((y - (-51.2f)) / 0.1f);
  gx = min(max(gx, 0), 1023);
  gy = min(max(gy, 0), 1023);
  const int cell = valid ? (gy * 1024 + gx) : -1;
  s_cell[t] = cell;
  __syncthreads();
  bool pos = valid;  // first-write-wins: invalidated if an earlier box owns this cell
  if (valid) {
    for (int j = 0; j < t; ++j)
      if (s_cell[j] == cell) { pos = false; break; }
  }
  float cs = 0.f, bs = 0.f, os = 0.f;
  if (pos) {
    const size_t base = (size_t)b * NCELLS + (size_t)cell;
    os = softplusf(-obj[base]);                      // BCE positive term
    const float* cl = cls + base * NCLS;             // CE w/ label smoothing
    float mx = cl[0];
    for (int i = 1; i < NCLS; ++i) mx = fmaxf(mx, cl[i]);
    float se = 0.f, ssum = 0.f;
    for (int i = 0; i < NCLS; ++i) { ssum += cl[i]; se += expf(cl[i] - mx); }
    const float lse = mx + logf(se);
    const float nll = lse - cl[lab];
    cs = 0.9f * nll + 0.1f * (lse - 0.1f * ssum);
    const float* bp = boxp + base * BD;              // smooth-L1 (beta=1)
    for (int i = 0; i < BD; ++i) {
      float d = bp[i] - box[i];
      float ad = fabsf(d);
      bs += (ad < 1.f) ? 0.5f * d * d : (ad - 0.5f);
    }
  }
  s_pos[t] = pos ? 1 : 0;
  s_c[t] = cs; s_b[t] = bs; s_o[t] = os;
  __syncthreads();
  int* pcell = (int*)(ws + OFF_POSCELL);
  if (pos) {  // ordered compaction (deterministic list order by box index)
    int slot = 0;
    for (int j = 0; j < t; ++j) slot += s_pos[j];
    pcell[b * NBOX + slot] = cell;
  }
  if (t == 0) {  // serial fixed-order reduction -> deterministic floats
    int cnt = 0; float C = 0.f, Bx = 0.f, O = 0.f;
    for (int j = 0; j < NBOX; ++j) { cnt += s_pos[j]; C += s_c[j]; Bx += s_b[j]; O += s_o[j]; }
    ((int*)(ws + OFF_POSCNT))[b] = cnt;
    float* bsums = (float*)(ws + OFF_BSUMS);
    bsums[b] = C; bsums[4 + b] = Bx; bsums[8 + b] = O;
  }
}

// ---------------- K3: pick level 0 (also computes P, k; excludes positives) ----------------
__global__ void k_pick0(const float* __restrict__ obj, uint8_t* __restrict__ ws) {
  uint32_t* h = (uint32_t*)(ws + OFF_HIST0);
  uint32_t* scal = (uint32_t*)(ws + OFF_SCAL);
  const int* pc = (const int*)(ws + OFF_POSCNT);
  const int* pcell = (const int*)(ws + OFF_POSCELL);
  int P = pc[0] + pc[1] + pc[2] + pc[3];
  for (int b = 0; b < BB; ++b)
    for (int i = 0; i < pc[b]; ++i) {
      uint32_t u = fkey(obj[(size_t)b * NCELLS + pcell[b * NBOX + i]]);
      h[u >> 20] -= 1u;
    }
  long long num_neg = (long long)TOTAL - P;
  long long kk = 3LL * (long long)(P > 1 ? P : 1);   // floor(3*max(P,1))
  if (kk < 1) kk = 1;
  if (kk > num_neg) kk = num_neg;
  if (P == 0) kk = num_neg;
  long long cum = 0; uint32_t T0 = 0, r0 = (uint32_t)kk;
  for (int bin = 4095; bin >= 0; --bin) {
    long long c = (long long)h[bin];
    if (cum + c >= kk) { T0 = (uint32_t)bin; r0 = (uint32_t)(kk - cum); break; }
    cum += c;
  }
  scal[0] = (uint32_t)P; scal[1] = (uint32_t)kk; scal[2] = T0; scal[3] = r0;
}

// ---------------- K4: histogram level 1 (bits 19:8 within T0 prefix) ----------------
__global__ void k_hist1(const float4* __restrict__ obj4, uint8_t* __restrict__ ws) {
  __shared__ uint32_t sh[4096];
  for (int i = threadIdx.x; i < 4096; i += THR) sh[i] = 0u;
  __syncthreads();
  const uint32_t T0 = ((const uint32_t*)(ws + OFF_SCAL))[2];
  const int stride = gridDim.x * blockDim.x;
  const int nvec = TOTAL / 4;
  for (int i = blockIdx.x * blockDim.x + threadIdx.x; i < nvec; i += stride) {
    __builtin_prefetch(obj4 + i + stride, 0, 0);
    float4 v = obj4[i];
    uint32_t u;
    u = fkey(v.x); if ((u >> 20) == T0) atomicAdd(&sh[(u >> 8) & 0xFFFu], 1u);
    u = fkey(v.y); if ((u >> 20) == T0) atomicAdd(&sh[(u >> 8) & 0xFFFu], 1u);
    u = fkey(v.z); if ((u >> 20) == T0) atomicAdd(&sh[(u >> 8) & 0xFFFu], 1u);
    u = fkey(v.w); if ((u >> 20) == T0) atomicAdd(&sh[(u >> 8) & 0xFFFu], 1u);
  }
  __syncthreads();
  uint32_t* gh = (uint32_t*)(ws + OFF_HIST1);
  for (int i = threadIdx.x; i < 4096; i += THR) {
    uint32_t c = sh[i];
    if (c) atomicAdd(&gh[i], c);
  }
}

// ---------------- K5: pick level 1 ----------------
__global__ void k_pick1(const float* __restrict__ obj, uint8_t* __restrict__ ws) {
  uint32_t* h = (uint32_t*)(ws + OFF_HIST1);
  uint32_t* scal = (uint32_t*)(ws + OFF_SCAL);
  const int* pc = (const int*)(ws + OFF_POSCNT);
  const int* pcell = (const int*)(ws + OFF_POSCELL);
  const uint32_t T0 = scal[2], r0 = scal[3];
  for (int b = 0; b < BB; ++b)
    for (int i = 0; i < pc[b]; ++i) {
      uint32_t u = fkey(obj[(size_t)b * NCELLS + pcell[b * NBOX + i]]);
      if ((u >> 20) == T0) h[(u >> 8) & 0xFFFu] -= 1u;
    }
  uint32_t cum = 0, T1 = 0, r1 = r0;
  for (int bin = 4095; bin >= 0; --bin) {
    uint32_t c = h[bin];
    if (cum + c >= r0) { T1 = (uint32_t)bin; r1 = r0 - cum; break; }
    cum += c;
  }
  scal[4] = (T0 << 12) | T1;   // 24-bit prefix
  scal[5] = r1;
}

// ---------------- K6: histogram level 2 (low 8 bits within 24-bit prefix) ----------------
__global__ void k_hist2(const float4* __restrict__ obj4, uint8_t* __restrict__ ws) {
  __shared__ uint32_t sh[256];
  sh[threadIdx.x] = 0u;
  __syncthreads();
  const uint32_t P01 = ((const uint32_t*)(ws + OFF_SCAL))[4];
  const int stride = gridDim.x * blockDim.x;
  const int nvec = TOTAL / 4;
  for (int i = blockIdx.x * blockDim.x + threadIdx.x; i < nvec; i += stride) {
    __builtin_prefetch(obj4 + i + stride, 0, 0);
    float4 v = obj4[i];
    uint32_t u;
    u = fkey(v.x); if ((u >> 8) == P01) atomicAdd(&sh[u & 0xFFu], 1u);
    u = fkey(v.y); if ((u >> 8) == P01) atomicAdd(&sh[u & 0xFFu], 1u);
    u = fkey(v.z); if ((u >> 8) == P01) atomicAdd(&sh[u & 0xFFu], 1u);
    u = fkey(v.w); if ((u >> 8) == P01) atomicAdd(&sh[u & 0xFFu], 1u);
  }
  __syncthreads();
  uint32_t* gh = (uint32_t*)(ws + OFF_HIST2);
  uint32_t c = sh[threadIdx.x];
  if (c) atomicAdd(&gh[threadIdx.x], c);
}

// ---------------- K7: pick level 2 -> exact threshold key u_star and tie count r ----------------
__global__ void k_pick2(const float* __restrict__ obj, uint8_t* __restrict__ ws) {
  uint32_t* h = (uint32_t*)(ws + OFF_HIST2);
  uint32_t* scal = (uint32_t*)(ws + OFF_SCAL);
  const int* pc = (const int*)(ws + OFF_POSCNT);
  const int* pcell = (const int*)(ws + OFF_POSCELL);
  const uint32_t P01 = scal[4], r1 = scal[5];
  for (int b = 0; b < BB; ++b)
    for (int i = 0; i < pc[b]; ++i) {
      uint32_t u = fkey(obj[(size_t)b * NCELLS + pcell[b * NBOX + i]]);
      if ((u >> 8) == P01) h[u & 0xFFu] -= 1u;
    }
  uint32_t cum = 0, T2 = 0, r2 = r1;
  for (int bin = 255; bin >= 0; --bin) {
    uint32_t c = h[bin];
    if (cum + c >= r1) { T2 = (uint32_t)bin; r2 = r1 - cum; break; }
    cum += c;
  }
  scal[6] = (P01 << 8) | T2;   // exact 32-bit threshold key
  scal[7] = r2;                // negatives to take at exactly u_star
}

// ---------------- K8: sum softplus over values strictly above threshold ----------------
__global__ void k_sum(const float4* __restrict__ obj4, uint8_t* __restrict__ ws) {
  const uint32_t us = ((const uint32_t*)(ws + OFF_SCAL))[6];
  const int stride = gridDim.x * blockDim.x;
  const int nvec = TOTAL / 4;
  float acc = 0.f;
  for (int i = blockIdx.x * blockDim.x + threadIdx.x; i < nvec; i += stride) {
    __builtin_prefetch(obj4 + i + stride, 0, 0);
    float4 v = obj4[i];
    if (fkey(v.x) > us) acc += softplusf(v.x);
    if (fkey(v.y) > us) acc += softplusf(v.y);
    if (fkey(v.z) > us) acc += softplusf(v.z);
    if (fkey(v.w) > us) acc += softplusf(v.w);
  }
  __shared__ float red[THR];
  red[threadIdx.x] = acc;
  __syncthreads();
  for (int s = THR / 2; s > 0; s >>= 1) {
    if (threadIdx.x < s) red[threadIdx.x] += red[threadIdx.x + s];
    __syncthreads();
  }
  if (threadIdx.x == 0) ((float*)(ws + OFF_PART))[blockIdx.x] = red[0];
}

// ---------------- K9: finalize (serial, deterministic) ----------------
__global__ void k_final(const float* __restrict__ obj, uint8_t* __restrict__ ws,
                        float* __restrict__ out) {
  const uint32_t* scal = (const uint32_t*)(ws + OFF_SCAL);
  const int P = (int)scal[0];
  const int kk = (int)scal[1];
  const uint32_t us = scal[6];
  const int r = (int)scal[7];
  const float* part = (const float*)(ws + OFF_PART);
  float s = 0.f;
  for (int i = 0; i < NBLK; ++i) s += part[i];
  // remove positive-cell contributions strictly above threshold
  const int* pc = (const int*)(ws + OFF_POSCNT);
  const int* pcell = (const int*)(ws + OFF_POSCELL);
  for (int b = 0; b < BB; ++b)
    for (int i = 0; i < pc[b]; ++i) {
      float x = obj[(size_t)b * NCELLS + pcell[b * NBOX + i]];
      if (fkey(x) > us) s -= softplusf(x);
    }
  // r tied negatives at exactly the threshold value
  s += (float)r * softplusf(fkey_inv(us));
  const float* bs = (const float*)(ws + OFF_BSUMS);
  const float clsS  = bs[0] + bs[1] + bs[2] + bs[3];
  const float boxS  = bs[4] + bs[5] + bs[6] + bs[7];
  const float objpS = bs[8] + bs[9] + bs[10] + bs[11];
  const float Pc = (float)(P > 1 ? P : 1);
  const float obj_pos = objpS / Pc;
  const float obj_neg = s / (float)(kk > 1 ? kk : 1);
  const float cls_loss = clsS / Pc;
  const int pd = P * BD;
  const float box_loss = boxS / (float)(pd > 1 ? pd : 1);
  const float obj_loss = obj_pos + obj_neg;
  out[0] = obj_pos + obj_neg + cls_loss + box_loss;  // all weights are 1.0
  out[1] = cls_loss;
  out[2] = box_loss;
  out[3] = obj_loss;
  out[4] = (float)P;
}

extern "C" void kernel_launch(void* const* d_in, const int* in_sizes, int n_in,
                              void* d_out, int out_size, void* d_ws, size_t ws_size,
                              hipStream_t stream) {
  const float* cls  = (const float*)d_in[0];
  const float* obj  = (const float*)d_in[1];
  const float* boxp = (const float*)d_in[2];
  const float* gtb  = (const float*)d_in[3];
  const int*   gtl  = (const int*)d_in[4];
  const float* gtm  = (const float*)d_in[5];
  float* out = (float*)d_out;
  uint8_t* ws = (uint8_t*)d_ws;

  const int nw = (int)(WS_BYTES / 4u);
  k_zero<<<(nw + THR - 1) / THR, THR, 0, stream>>>((uint32_t*)ws, nw);
  k_assign<<<BB, NBOX, 0, stream>>>(cls, obj, boxp, gtb, gtl, gtm, ws);
  k_hist0<<<TOTAL / (2 * TILE), THR, 0, stream>>>(obj, ws);   // 512 blocks, 2 tiles each
  k_pick0<<<1, 1, 0, stream>>>(obj, ws);
  k_hist1<<<NBLK, THR, 0, stream>>>((const float4*)obj, ws);
  k_pick1<<<1, 1, 0, stream>>>(obj, ws);
  k_hist2<<<NBLK, THR, 0, stream>>>((const float4*)obj, ws);
  k_pick2<<<1, 1, 0, stream>>>(obj, ws);
  k_sum<<<NBLK, THR, 0, stream>>>((const float4*)obj, ws);
  k_final<<<1, 1, 0, stream>>>(obj, ws, out);
}